// PLIFNode_13589276524701
// MI455X (gfx1250) — compile-verified
//
#include <hip/hip_runtime.h>
#include <math.h>

// PLIF forward for x:[B=32, T=8, C=128, H=32, W=32] f32, scalar w.
// Memory-bound streaming recurrence: 268 MB total traffic -> ~11.5 us floor
// at 23.3 TB/s. No GEMM structure (nonlinear reset per step), so the optimal
// kernel is a wave32-coalesced, 128-bit, NT-hinted stream with 8-deep MLP.

typedef float v4f __attribute__((ext_vector_type(4)));

#define PLIF_T      8
#define PLIF_B      32u
#define PLIF_CHW    (128u * 32u * 32u)        // 131072, divisible by 4
#define PLIF_NSPAT  (PLIF_B * PLIF_CHW)       // 4,194,304 spatial points
#define PLIF_NUNITS (PLIF_NSPAT / 4u)         // 1,048,576 float4 units
#define PLIF_VTH    0.5f

__global__ __launch_bounds__(256) void plif_fwd_kernel(
    const float* __restrict__ x,
    const float* __restrict__ w,
    float* __restrict__ out) {

  const unsigned u = blockIdx.x * blockDim.x + threadIdx.x;
  if (u >= PLIF_NUNITS) return;

  // Spatial flat id (b * CHW + r), vec4-aligned.
  const unsigned s4 = u * 4u;
  const unsigned b  = s4 / PLIF_CHW;
  const unsigned r  = s4 - b * PLIF_CHW;

  // tau = sigmoid(w): uniform scalar, one cached load per wave.
  const float wv  = *w;
  const float tau = 1.0f / (1.0f + __expf(-wv));

  // Issue all 8 strided time-step loads up front (8-deep LOADcnt pipeline).
  // Per wave32, each load covers 512 contiguous bytes -> fully coalesced.
  v4f xv[PLIF_T];
#pragma unroll
  for (int t = 0; t < PLIF_T; ++t) {
    const size_t off = (size_t)(b * (unsigned)PLIF_T + (unsigned)t) * PLIF_CHW + r;
    xv[t] = __builtin_nontemporal_load((const v4f*)(x + off));
  }

  // Sequential T=8 recurrence with hard reset; branchless per component.
  v4f mem = {0.0f, 0.0f, 0.0f, 0.0f};
#pragma unroll
  for (int t = 0; t < PLIF_T; ++t) {
    v4f sp;
#pragma unroll
    for (int c = 0; c < 4; ++c) {
      float m      = tau * mem[c] + xv[t][c];
      const bool f = (m > PLIF_VTH);       // surrogate_spike(mem - Vth) forward
      sp[c]        = f ? 1.0f : 0.0f;
      mem[c]       = f ? 0.0f : m;         // hard reset where spiked
    }
    const size_t off = (size_t)(b * (unsigned)PLIF_T + (unsigned)t) * PLIF_CHW + r;
    __builtin_nontemporal_store(sp, (v4f*)(out + off));  // single-touch stream
  }
}

extern "C" void kernel_launch(void* const* d_in, const int* in_sizes, int n_in,
                              void* d_out, int out_size, void* d_ws, size_t ws_size,
                              hipStream_t stream) {
  (void)in_sizes; (void)n_in; (void)out_size; (void)d_ws; (void)ws_size;
  const float* x = (const float*)d_in[0];
  const float* w = (const float*)d_in[1];
  float* out     = (float*)d_out;

  const unsigned threads = 256;
  const unsigned blocks  = (PLIF_NUNITS + threads - 1) / threads;  // 4096
  plif_fwd_kernel<<<blocks, threads, 0, stream>>>(x, w, out);
}